// DiversityInjection_5274219840043
// MI455X (gfx1250) — compile-verified
//
#include <hip/hip_runtime.h>
#include <math.h>

#define NUM_AGENTS 256
#define HIDDEN 1024
#define RANK 32
#define NTOK 8192
#define ALPHA_MAX 5.0f
#define LN_EPS 1e-5f
#define PERT_STRIDE 1028   // rows m and m+8 land in disjoint LDS bank ranges
#define V_STRIDE 1032      // rows k and k+2 differ by 16 banks; 16B-aligned rows

typedef __attribute__((ext_vector_type(2))) float v2f;
typedef __attribute__((ext_vector_type(8))) float v8f;

__global__ __launch_bounds__(128) void div_inject_kernel(
    const float* __restrict__ h,
    const float* __restrict__ log_alpha,
    const float* __restrict__ gamma,
    const float* __restrict__ beta,
    const float* __restrict__ pu,
    const float* __restrict__ pv,
    const int* __restrict__ agent_ids,
    float* __restrict__ out)
{
    __shared__ int   s_tok[NTOK];
    __shared__ int   s_count;
    __shared__ float s_inter[2][16][RANK];       // two K-partials of [16 x 32]
    __shared__ float s_pert[16][PERT_STRIDE];    // perturbation tile [16 x 1024]
    __shared__ float s_v[RANK][V_STRIDE];        // V[32 x 1024] staged once per agent

    const int agent = blockIdx.x;
    const int tid   = threadIdx.x;
    const int lane  = tid & 31;
    // Wave index forced scalar so wave-level loops compile as uniform/scalar loops.
    const int wv    = __builtin_amdgcn_readfirstlane(tid >> 5);

    const float* __restrict__ U = pu + (size_t)agent * HIDDEN * RANK; // [1024 x 32]
    const float* __restrict__ V = pv + (size_t)agent * RANK * HIDDEN; // [32 x 1024]

    // ---- Phase 0a: kick off async stage of V into LDS (ASYNCcnt-tracked),
    //      overlapped with the agent-id gather scan below. ----
    if (tid == 0) s_count = 0;
    __syncthreads();

    for (int r = wv; r < RANK; r += 4) {
        const float* src = V + (size_t)r * HIDDEN;
        for (int c = lane * 4; c < HIDDEN; c += 128) {
            unsigned ldsa = (unsigned)(size_t)&s_v[r][c];
            const float* g = src + c;
            asm volatile("global_load_async_to_lds_b128 %0, %1, off"
                         :: "v"(ldsa), "v"(g) : "memory");
        }
    }

    // ---- Phase 0b: gather token indices belonging to this agent ----
    for (int i = tid; i < NTOK; i += blockDim.x) {
        int id = agent_ids[i] & (NUM_AGENTS - 1);
        if (id == agent) {
            int p = atomicAdd(&s_count, 1);
            s_tok[p] = i;
        }
    }
    __syncthreads();
    const int count = s_count;
    if (count == 0) return;                      // uniform; S_ENDPGM waits idle
    const int padded = (count + 15) & ~15;
    for (int i = count + tid; i < padded; i += blockDim.x) s_tok[i] = s_tok[0];

    // Each wave drains its own async copies; the barrier at the top of the
    // tile loop then makes all of s_v visible workgroup-wide.
    asm volatile("s_wait_asynccnt 0" ::: "memory");

    const float alpha = fminf(expf(log_alpha[0]), ALPHA_MAX);

    // WMMA fp32 16x16x4 fragment geometry (ISA 7.12.2):
    //   A: lane holds row M = lane&15; VGPR0 = K (koff), VGPR1 = K+1 (koff = 2*(lane>=16))
    //   B: lane holds col N = lane&15; same K striping
    //   C/D: VGPR i -> row M = i + 8*(lane>=16), col N = lane&15
    const int mrow = lane & 15;
    const int koff = (lane >> 4) << 1;           // 0 or 2
    const int rhi  = (lane >> 4) << 3;           // 0 or 8

    for (int t0 = 0; t0 < padded; t0 += 16) {
        __syncthreads();   // prev epilogue done with s_pert; s_v/s_tok visible

        // ---- GEMM1: inter[16x32] = Htile[16x1024] * U[1024x32] ----
        {
            const int ntile = wv >> 1;                   // rank-column tile (0..1)
            const int kpart = wv & 1;                    // K split half
            const int n0    = ntile * 16;
            const int kbeg  = kpart * (HIDDEN / 2);
            const int kend  = kbeg + (HIDDEN / 2);
            const int tok   = s_tok[t0 + mrow];
            const float* hrow = h + (size_t)tok * HIDDEN;

            v8f acc = {};
            for (int k = kbeg; k < kend; k += 4) {
                v2f a, b;
                const float2 av = *reinterpret_cast<const float2*>(hrow + k + koff);
                a.x = av.x; a.y = av.y;
                b.x = U[(size_t)(k + koff)     * RANK + n0 + mrow];
                b.y = U[(size_t)(k + koff + 1) * RANK + n0 + mrow];
                acc = __builtin_amdgcn_wmma_f32_16x16x4_f32(
                          false, a, false, b, (short)0, acc, false, false);
            }
            #pragma unroll
            for (int i = 0; i < 8; ++i)
                s_inter[kpart][i + rhi][n0 + mrow] = acc[i];
        }
        __syncthreads();

        // ---- GEMM2: pert[16x1024] = inter[16x32] * V_lds[32x1024] ----
        {
            v2f afrag[8];                                // A reused across all N-tiles
            #pragma unroll
            for (int ks = 0; ks < 8; ++ks) {
                const int k0 = ks * 4 + koff;
                afrag[ks].x = s_inter[0][mrow][k0]     + s_inter[1][mrow][k0];
                afrag[ks].y = s_inter[0][mrow][k0 + 1] + s_inter[1][mrow][k0 + 1];
            }
            for (int nt = wv; nt < HIDDEN / 16; nt += 4) {
                const int n0 = nt * 16;
                v8f acc = {};
                #pragma unroll
                for (int ks = 0; ks < 8; ++ks) {
                    const int k0 = ks * 4 + koff;
                    v2f b;
                    b.x = s_v[k0][n0 + mrow];
                    b.y = s_v[k0 + 1][n0 + mrow];
                    acc = __builtin_amdgcn_wmma_f32_16x16x4_f32(
                              false, afrag[ks], false, b, (short)0, acc, false, false);
                }
                #pragma unroll
                for (int i = 0; i < 8; ++i)
                    s_pert[i + rhi][n0 + mrow] = acc[i];
            }
        }
        __syncthreads();

        // ---- Epilogue: out = LayerNorm(h + alpha * pert) per token row ----
        for (int r = 0; r < 4; ++r) {
            const int m = wv * 4 + r;                    // scalar per wave
            if (t0 + m >= count) break;                  // skip padding rows
            const int tok = s_tok[t0 + m];
            const float* hrow = h + (size_t)tok * HIDDEN;
            float sum = 0.f, sumsq = 0.f;
            for (int j = lane; j < HIDDEN; j += 32) {
                const float x = hrow[j] + alpha * s_pert[m][j];
                sum += x; sumsq += x * x;
            }
            #pragma unroll
            for (int off = 16; off > 0; off >>= 1) {
                sum   += __shfl_xor(sum,   off, 32);
                sumsq += __shfl_xor(sumsq, off, 32);
            }
            const float mean = sum * (1.0f / HIDDEN);
            const float var  = sumsq * (1.0f / HIDDEN) - mean * mean;
            const float rstd = rsqrtf(var + LN_EPS);
            float* orow = out + (size_t)tok * HIDDEN;
            for (int j = lane; j < HIDDEN; j += 32) {
                const float x = hrow[j] + alpha * s_pert[m][j];
                orow[j] = (x - mean) * rstd * gamma[j] + beta[j];
            }
        }
    }
}

extern "C" void kernel_launch(void* const* d_in, const int* in_sizes, int n_in,
                              void* d_out, int out_size, void* d_ws, size_t ws_size,
                              hipStream_t stream) {
    const float* h         = (const float*)d_in[0];
    const float* log_alpha = (const float*)d_in[1];
    const float* gamma     = (const float*)d_in[2];
    const float* beta      = (const float*)d_in[3];
    const float* pu        = (const float*)d_in[4];
    const float* pv        = (const float*)d_in[5];
    const int*   ids       = (const int*)d_in[6];
    float*       out       = (float*)d_out;

    div_inject_kernel<<<NUM_AGENTS, 128, 0, stream>>>(
        h, log_alpha, gamma, beta, pu, pv, ids, out);
}